// RegionProposalNetwork_17686675325261
// MI455X (gfx1250) — compile-verified
//
#include <hip/hip_runtime.h>
#include <math.h>

typedef __attribute__((ext_vector_type(2))) float v2f;
typedef __attribute__((ext_vector_type(8))) float v8f;

#define HW     4096
#define NA     9
#define NPROP  (HW * NA)     // 36864
#define CIN    512
#define NIMG   8
#define NPRE   12000
#define NSLOT  12288         // NPRE rounded to 12*1024 slots
#define NPOST  600
#define NEGINF (-__builtin_inff())

// ------------------------------------------------------------------
// Kernel 1: fused 1x1-conv GEMM via f32 WMMA (M=54 padded to 64)
//   block = 128 threads (4 wave32), each wave owns one 16-row M tile
//   grid  = (256 hw-tiles, 8 images)
// ------------------------------------------------------------------
__global__ __launch_bounds__(128) void rpn_gemm_wmma(
    const float* __restrict__ x,
    const float* __restrict__ w_score, const float* __restrict__ b_score,
    const float* __restrict__ w_loc,   const float* __restrict__ b_loc,
    float* __restrict__ out_locs,      float* __restrict__ out_scores)
{
  const int img  = blockIdx.y;
  const int hw0  = blockIdx.x << 4;
  const int lane = threadIdx.x & 31;
  const int o0   = (threadIdx.x >> 5) << 4;  // 0,16,32,48
  const int col  = lane & 15;
  const int half = lane >> 4;

  // B-fragment source: x[img][c][hw0+col], c-stride = HW
  const float* __restrict__ xb = x + (size_t)img * CIN * HW + hw0 + col;
  // A-fragment source: combined [loc(36) ; score(18)] weight row
  const int orow = o0 + col;
  const float* wrow = (orow < 36) ? (w_loc + (size_t)orow * CIN)
                    : (orow < 54) ? (w_score + (size_t)(orow - 36) * CIN)
                                  : nullptr;
  __builtin_prefetch(xb, 0, 1);              // global_prefetch_b8
  if (wrow) __builtin_prefetch(wrow, 0, 3);

  v8f acc = {};
  #pragma unroll 4
  for (int k0 = 0; k0 < CIN; k0 += 4) {
    const int ka = k0 + (half << 1);
    v2f a, b;
    if (wrow) { a.x = wrow[ka]; a.y = wrow[ka + 1]; }
    else      { a.x = 0.f;      a.y = 0.f; }
    b.x = xb[(size_t)ka * HW];
    b.y = xb[(size_t)(ka + 1) * HW];
    // D = A(16x4) * B(4x16) + C, fp32 all the way (matches reference)
    acc = __builtin_amdgcn_wmma_f32_16x16x4_f32(false, a, false, b,
                                                (short)0, acc, false, false);
  }

  // C layout: VGPR r -> M = r + 8*half, N = col.  Scatter into the
  // (n, hw*9 + a, {4|2}) permuted layout, folding in the bias.
  const int hw = hw0 + col;
  const size_t pbase = (size_t)img * NPROP + (size_t)hw * NA;
  #pragma unroll
  for (int r = 0; r < 8; ++r) {
    const int o = o0 + r + (half << 3);
    if (o >= 54) continue;
    const float v = acc[r];
    if (o < 36) {
      out_locs[(pbase + (o >> 2)) * 4 + (o & 3)] = v + b_loc[o];
    } else {
      const int oo = o - 36;
      out_scores[(pbase + (oo >> 1)) * 2 + (oo & 1)] = v + b_score[oo];
    }
  }
}

// ------------------------------------------------------------------
// Kernel 2: per-proposal anchor gen + box decode + clip + fg softmax
// ------------------------------------------------------------------
__global__ __launch_bounds__(256) void rpn_decode(
    const float* __restrict__ locs, const float* __restrict__ scores,
    float* __restrict__ ws_boxes, float* __restrict__ ws_scores,
    float* __restrict__ out_anchor)
{
  const int idx = blockIdx.x * 256 + threadIdx.x;   // 0 .. 8*36864-1
  const int img = idx / NPROP;
  const int p   = idx - img * NPROP;
  const int a   = p % NA;
  const int hw  = p / NA;
  const float sx = (float)(hw & 63) * 16.f;
  const float sy = (float)(hw >> 6) * 16.f;
  const float ratios[3] = {0.5f, 1.f, 2.f};
  const float scales[3] = {8.f, 16.f, 32.f};
  const float r  = ratios[a / 3];
  const float s  = scales[a % 3];
  const float hA = 16.f * s * sqrtf(r);
  const float wA = 16.f * s * sqrtf(1.f / r);

  if (img == 0) {
    out_anchor[p * 4 + 0] = sx - 0.5f * wA;
    out_anchor[p * 4 + 1] = sy - 0.5f * hA;
    out_anchor[p * 4 + 2] = sx + 0.5f * wA;
    out_anchor[p * 4 + 3] = sy + 0.5f * hA;
  }

  const float l0 = locs[(size_t)idx * 4 + 0];
  const float l1 = locs[(size_t)idx * 4 + 1];
  const float l2 = locs[(size_t)idx * 4 + 2];
  const float l3 = locs[(size_t)idx * 4 + 3];
  const float cx = l0 * wA + sx;
  const float cy = l1 * hA + sy;
  const float bw = expf(l2) * wA;
  const float bh = expf(l3) * hA;
  const float x1 = fminf(fmaxf(cx - 0.5f * bw, 0.f), 1024.f);
  const float y1 = fminf(fmaxf(cy - 0.5f * bh, 0.f), 1024.f);
  const float x2 = fminf(fmaxf(cx + 0.5f * bw, 0.f), 1024.f);
  const float y2 = fminf(fmaxf(cy + 0.5f * bh, 0.f), 1024.f);
  ws_boxes[(size_t)idx * 4 + 0] = x1;
  ws_boxes[(size_t)idx * 4 + 1] = y1;
  ws_boxes[(size_t)idx * 4 + 2] = x2;
  ws_boxes[(size_t)idx * 4 + 3] = y2;

  const bool valid = ((x2 - x1) >= 16.f) && ((y2 - y1) >= 16.f);
  const float s0 = scores[(size_t)idx * 2 + 0];
  const float s1 = scores[(size_t)idx * 2 + 1];
  const float fg = 1.f / (1.f + expf(s0 - s1));  // softmax fg prob
  ws_scores[idx] = valid ? fg : NEGINF;
}

// ------------------------------------------------------------------
// Kernel 3: per-image pre-NMS top-k (via score threshold) + greedy NMS
//   1 block (1024 threads = 32 wave32) per image
// ------------------------------------------------------------------
__global__ __launch_bounds__(1024) void rpn_nms(
    const float* __restrict__ ws_boxes, const float* __restrict__ ws_scores,
    float* __restrict__ ws_cbox, float* __restrict__ ws_cscore,
    float* __restrict__ out_rois)
{
  const int img = blockIdx.x;
  const int tid = threadIdx.x;
  const float* S = ws_scores + (size_t)img * NPROP;
  const float* B = ws_boxes  + (size_t)img * NPROP * 4;
  float* CB = ws_cbox   + (size_t)img * NSLOT * 4;
  float* CS = ws_cscore + (size_t)img * NSLOT;

  __shared__ float redV[32];
  __shared__ int   redI[32];
  __shared__ int   s_cnt;
  __shared__ int   s_bslot;
  __shared__ float s_best[5];   // x1,y1,x2,y2,score

  // ---- Phase A: binary-search score threshold ~= N_PRE'th score ----
  float lo = 0.f, hi = 1.f;     // fg scores are sigmoids in (0,1)
  for (int it = 0; it < 24; ++it) {
    const float mid = 0.5f * (lo + hi);
    int cnt = 0;
    for (int p = tid; p < NPROP; p += 1024) cnt += (S[p] > mid) ? 1 : 0;
    for (int off = 16; off; off >>= 1) cnt += __shfl_down(cnt, off, 32);
    if ((tid & 31) == 0) redI[tid >> 5] = cnt;
    __syncthreads();
    if (tid < 32) {
      int c = redI[tid];
      for (int off = 16; off; off >>= 1) c += __shfl_down(c, off, 32);
      if (tid == 0) redI[0] = c;
    }
    __syncthreads();
    const int total = redI[0];
    if (total > NPRE) lo = mid; else hi = mid;
    __syncthreads();
  }
  const float thresh = lo;      // count(>lo) >= NPRE (capped at NSLOT below)

  // ---- Phase B: compact survivors into workspace ----
  if (tid == 0) s_cnt = 0;
  __syncthreads();
  for (int p = tid; p < NPROP; p += 1024) {
    const float sc = S[p];
    if (sc > thresh) {
      const int slot = atomicAdd(&s_cnt, 1);
      if (slot < NSLOT) {
        CS[slot] = sc;
        CB[slot * 4 + 0] = B[(size_t)p * 4 + 0];
        CB[slot * 4 + 1] = B[(size_t)p * 4 + 1];
        CB[slot * 4 + 2] = B[(size_t)p * 4 + 2];
        CB[slot * 4 + 3] = B[(size_t)p * 4 + 3];
      }
    }
  }
  __threadfence_block();
  __syncthreads();
  const int total = (s_cnt < NSLOT) ? s_cnt : NSLOT;

  // ---- Phase C: survivors to registers (12 slots/thread) ----
  float  myS[12];
  float4 myB[12];
  #pragma unroll
  for (int k = 0; k < 12; ++k) {
    const int i = tid + (k << 10);
    if (i < total) {
      myS[k] = CS[i];
      myB[k] = make_float4(CB[i*4+0], CB[i*4+1], CB[i*4+2], CB[i*4+3]);
    } else {
      myS[k] = NEGINF;
      myB[k] = make_float4(-1.e4f, -1.e4f, -1.e4f, -1.e4f);
    }
  }

  // ---- Phase D: greedy NMS, 600 sequential picks ----
  float* outR = out_rois + (size_t)img * NPOST * 4;
  int pads = 0;
  for (int it = 0; it < NPOST; ++it) {
    // local argmax over own 12 slots
    float bv = NEGINF; int bk = 0;
    #pragma unroll
    for (int k = 0; k < 12; ++k) if (myS[k] > bv) { bv = myS[k]; bk = k; }
    int bslot = tid + (bk << 10);
    // wave32 argmax
    for (int off = 16; off; off >>= 1) {
      const float ov = __shfl_down(bv, off, 32);
      const int   os = __shfl_down(bslot, off, 32);
      if (ov > bv) { bv = ov; bslot = os; }
    }
    if ((tid & 31) == 0) { redV[tid >> 5] = bv; redI[tid >> 5] = bslot; }
    __syncthreads();
    if (tid < 32) {
      float v = redV[tid]; int sl = redI[tid];
      for (int off = 16; off; off >>= 1) {
        const float ov = __shfl_down(v, off, 32);
        const int   os = __shfl_down(sl, off, 32);
        if (ov > v) { v = ov; sl = os; }
      }
      if (tid == 0) { s_bslot = sl; s_best[4] = v; }
    }
    __syncthreads();
    const int  best     = s_bslot;
    const bool validSel = s_best[4] > -1.e30f;
    if (validSel && (best & 1023) == tid) {
      const int k = best >> 10;
      s_best[0] = myB[k].x; s_best[1] = myB[k].y;
      s_best[2] = myB[k].z; s_best[3] = myB[k].w;
      myS[k] = NEGINF;
    }
    __syncthreads();
    const float bx1 = s_best[0], by1 = s_best[1];
    const float bx2 = s_best[2], by2 = s_best[3];
    if (validSel) {
      const float areaB = fmaxf(bx2 - bx1, 0.f) * fmaxf(by2 - by1, 0.f);
      #pragma unroll
      for (int k = 0; k < 12; ++k) {
        const float ix1 = fmaxf(bx1, myB[k].x);
        const float iy1 = fmaxf(by1, myB[k].y);
        const float ix2 = fminf(bx2, myB[k].z);
        const float iy2 = fminf(by2, myB[k].w);
        const float inter = fmaxf(ix2 - ix1, 0.f) * fmaxf(iy2 - iy1, 0.f);
        const float areaK = fmaxf(myB[k].z - myB[k].x, 0.f) *
                            fmaxf(myB[k].w - myB[k].y, 0.f);
        const float iou = inter / (areaB + areaK - inter + 1e-9f);
        if (iou > 0.7f) myS[k] = NEGINF;
      }
    }
    if (tid == 0) {
      if (validSel) {
        outR[it * 4 + 0] = bx1; outR[it * 4 + 1] = by1;
        outR[it * 4 + 2] = bx2; outR[it * 4 + 3] = by2;
      } else {
        outR[it * 4 + 0] = CB[pads * 4 + 0];
        outR[it * 4 + 1] = CB[pads * 4 + 1];
        outR[it * 4 + 2] = CB[pads * 4 + 2];
        outR[it * 4 + 3] = CB[pads * 4 + 3];
        ++pads;
      }
    }
    __syncthreads();
  }
}

// ------------------------------------------------------------------
extern "C" void kernel_launch(void* const* d_in, const int* in_sizes, int n_in,
                              void* d_out, int out_size, void* d_ws, size_t ws_size,
                              hipStream_t stream) {
  (void)in_sizes; (void)n_in; (void)out_size; (void)ws_size;
  const float* x       = (const float*)d_in[0];
  const float* w_score = (const float*)d_in[1];
  const float* b_score = (const float*)d_in[2];
  const float* w_loc   = (const float*)d_in[3];
  const float* b_loc   = (const float*)d_in[4];

  float* out        = (float*)d_out;
  float* out_locs   = out;                                 // 8*36864*4
  float* out_scores = out + (size_t)NIMG * NPROP * 4;      // 8*36864*2
  float* out_rois   = out + (size_t)NIMG * NPROP * 6;      // 8*600*4
  float* out_anch   = out_rois + (size_t)NIMG * NPOST * 4; // 1*36864*4

  float* ws        = (float*)d_ws;
  float* ws_boxes  = ws;                                   // 8*36864*4
  float* ws_scores = ws_boxes  + (size_t)NIMG * NPROP * 4; // 8*36864
  float* ws_cbox   = ws_scores + (size_t)NIMG * NPROP;     // 8*12288*4
  float* ws_cscore = ws_cbox   + (size_t)NIMG * NSLOT * 4; // 8*12288

  rpn_gemm_wmma<<<dim3(HW / 16, NIMG), 128, 0, stream>>>(
      x, w_score, b_score, w_loc, b_loc, out_locs, out_scores);

  rpn_decode<<<(NIMG * NPROP) / 256, 256, 0, stream>>>(
      out_locs, out_scores, ws_boxes, ws_scores, out_anch);

  rpn_nms<<<NIMG, 1024, 0, stream>>>(
      ws_boxes, ws_scores, ws_cbox, ws_cscore, out_rois);
}